// LSTMImplementation_10771777978481
// MI455X (gfx1250) — compile-verified
//
#include <hip/hip_runtime.h>
#include <math.h>

typedef __attribute__((ext_vector_type(2))) float v2f;
typedef __attribute__((ext_vector_type(8))) float v8f;

#define S_LEN   512
#define BATCH   64
#define INP     512
#define HID     512
#define KTOT    1024           // INP + HID
#define NBLK_DIR 32            // HID / 16 columns per block
#define THREADS 256

// ---- LDS layout (floats) ----
#define WL_STRIDE  65
#define WL_FLOATS  (KTOT * WL_STRIDE)      // 66560 : weights [k][q] (q = 64 gate rows)
#define ACT_STRIDE 65
#define ACT_FLOATS (64 * ACT_STRIDE)       // 4160  : one activation chunk, k-major [kk][m]
#define GLD_STRIDE 65                      // gate dump aliases act buffer 0
#define CST_FLOATS (16 * 64)               // 1024  : per-block state [j][b]
#define LDS_FLOATS (WL_FLOATS + 2 * ACT_FLOATS + 3 * CST_FLOATS + 64 + 64)
#define LDS_BYTES  (LDS_FLOATS * 4)        // 312,320 B  (<= 320 KB WGP LDS)

__device__ __forceinline__ float sigm(float x) { return 1.0f / (1.0f + expf(-x)); }

__device__ __forceinline__ void wait_async() {
#if __has_builtin(__builtin_amdgcn_s_wait_asynccnt)
  __builtin_amdgcn_s_wait_asynccnt(0);
#else
  asm volatile("s_wait_asynccnt 0x0" ::: "memory");
#endif
}

// Async global->LDS copy of one 64(m) x 64(k) fp32 chunk, transposed to
// k-major [k][m] in LDS via per-lane addressing. Global side: 32 lanes read
// 32 consecutive k for fixed m -> fully coalesced 128B per instruction.
// Tracked by ASYNCcnt (no VGPR round-trip, overlaps WMMA of previous chunk).
__device__ __forceinline__ void issue_chunk_async(const float* __restrict__ src,
                                                  unsigned abase_bytes,
                                                  int wv, int lane) {
  unsigned long long gbase = (unsigned long long)(size_t)src;
  #pragma unroll
  for (int r = 0; r < 16; ++r) {
    int e = wv * 512 + r * 32 + lane;     // 0..4095 over 8 waves
    int k = e & 63, m = e >> 6;
    unsigned goff = (unsigned)((m * INP + k) * 4);
    unsigned loff = abase_bytes + (unsigned)((k * ACT_STRIDE + m) * 4);
    asm volatile("global_load_async_to_lds_b32 %0, %1, %2"
                 :: "v"(loff), "v"(goff), "s"(gbase)
                 : "memory");
  }
}

// D = A(16x4) * B(4x16) + C, fp32 WMMA. Fallback: exact cross-lane emulation.
__device__ __forceinline__ v8f wmma4(v2f a, v2f b, v8f c, int lane) {
#if __has_builtin(__builtin_amdgcn_wmma_f32_16x16x4_f32)
  (void)lane;
  return __builtin_amdgcn_wmma_f32_16x16x4_f32(false, a, false, b, (short)0, c,
                                               false, false);
#else
  const bool hi = lane >= 16;
  const int n = lane & 15;
  #pragma unroll
  for (int v = 0; v < 8; ++v) {
    int m = v + (hi ? 8 : 0);
    float acc = c[v];
    #pragma unroll
    for (int k = 0; k < 4; ++k) {
      int src = (m & 15) + ((k >> 1) << 4);
      float av = __shfl((k & 1) ? a.y : a.x, src, 32);
      int srcb = n + ((k >> 1) << 4);
      float bv = __shfl((k & 1) ? b.y : b.x, srcb, 32);
      acc = fmaf(av, bv, acc);
    }
    c[v] = acc;
  }
  return c;
#endif
}

__device__ __forceinline__ void compute_chunk(const float* __restrict__ Wl,
                                              const float* __restrict__ abuf,
                                              int kbase, int N0, int M0a, int M0b,
                                              int l, bool hi, int lane,
                                              v8f& accA, v8f& accB) {
  #pragma unroll
  for (int ks = 0; ks < 16; ++ks) {
    const int k0 = ks * 4 + (hi ? 2 : 0);
    v2f bf, aA, aB;
    bf.x = Wl[(kbase + k0) * WL_STRIDE + N0 + l];
    bf.y = Wl[(kbase + k0 + 1) * WL_STRIDE + N0 + l];
    aA.x = abuf[k0 * ACT_STRIDE + M0a + l];
    aA.y = abuf[(k0 + 1) * ACT_STRIDE + M0a + l];
    aB.x = abuf[k0 * ACT_STRIDE + M0b + l];
    aB.y = abuf[(k0 + 1) * ACT_STRIDE + M0b + l];
    accA = wmma4(aA, bf, accA, lane);
    accB = wmma4(aB, bf, accB, lane);
  }
}

__global__ void lstm_init(const float* __restrict__ h0g, float* hbuf,
                          unsigned* cnt) {
  int idx = blockIdx.x * blockDim.x + threadIdx.x;   // 0 .. 2*64*512-1
  hbuf[idx] = h0g[idx];                              // hbuf[0][d][b][h]
  if (idx < 2) cnt[idx] = 0u;
}

__global__ void __launch_bounds__(THREADS, 1)
lstm_persist(const float* __restrict__ X, const float* __restrict__ W,
             const float* __restrict__ R, const float* __restrict__ Bg,
             const int* __restrict__ seq, const float* __restrict__ h0g,
             const float* __restrict__ c0g, const float* __restrict__ P,
             float* __restrict__ Y, float* hbuf, unsigned* cnt) {
  extern __shared__ float lds[];
  float* Wl   = lds;                         // [k][q], stride 65
  float* act0 = Wl + WL_FLOATS;              // chunk buffer 0 (aliased by gld)
  float* act1 = act0 + ACT_FLOATS;           // chunk buffer 1
  float* gld  = act0;                        // gate dump [m][q] (post-K-loop only)
  float* cst  = act1 + ACT_FLOATS;           // [j][b]
  float* h0s  = cst + CST_FLOATS;            // [j][b]
  float* c0s  = h0s + CST_FLOATS;            // [j][b]
  float* Ps   = c0s + CST_FLOATS;            // 48 used (i|f|o), pad to 64
  int*   sls  = (int*)(Ps + 64);             // 64

  const int d   = blockIdx.x >> 5;           // direction
  const int j0  = (blockIdx.x & 31) * 16;    // h-column base
  const int tid = threadIdx.x;

  // ---- one-time preload: weights (block's 64 gate rows x K=1024), LDS-resident ----
  // gate row order per block: q/16 -> {i, o, f, c~} at 4H offsets {0,H,2H,3H}
  for (int q = 0; q < 64; ++q) {
    int grow = (q >> 4) * HID + j0 + (q & 15);
    const float* wrow = W + (d * 4 * HID + grow) * INP;
    const float* rrow = R + (d * 4 * HID + grow) * HID;
    for (int r = tid; r < INP; r += THREADS) Wl[r * WL_STRIDE + q] = wrow[r];
    for (int r = tid; r < HID; r += THREADS) Wl[(INP + r) * WL_STRIDE + q] = rrow[r];
  }
  for (int e = tid; e < 16 * 64; e += THREADS) {  // state: [j][b]
    int j = e >> 6, b = e & 63;
    float cv = c0g[(d * BATCH + b) * HID + j0 + j];
    cst[e] = cv; c0s[e] = cv;
    h0s[e] = h0g[(d * BATCH + b) * HID + j0 + j];
  }
  if (tid < 48) Ps[tid] = P[d * 3 * HID + (tid >> 4) * HID + j0 + (tid & 15)];
  if (tid < 64) sls[tid] = seq[tid];
  __syncthreads();

  const int wv = tid >> 5, lane = tid & 31, l = lane & 15;
  const bool hi = lane >= 16;
  const int N0 = (wv & 3) * 16;              // gate-row tile
  const int M0a = (wv >> 2) * 32, M0b = M0a + 16;  // batch tiles
  const unsigned a0off = (unsigned)((const char*)act0 - (const char*)lds);
  const unsigned a1off = (unsigned)((const char*)act1 - (const char*)lds);

  // per-lane bias (Wb + Rb) for this wave's 16 gate rows
  const int qn = N0 + l;
  const int growb = (qn >> 4) * HID + j0 + (qn & 15);
  const float bsum = Bg[d * 8 * HID + growb] + Bg[d * 8 * HID + 4 * HID + growb];

  int cur = 0;
  // prime: async-load chunk 0 (pure X) of the first timestep into act0
  {
    const int t0 = (d == 0) ? 0 : (S_LEN - 1);
    issue_chunk_async(X + (size_t)t0 * BATCH * INP, a0off, wv, lane);
  }

  for (int s = 0; s < S_LEN; ++s) {
    const int t = (d == 0) ? s : (S_LEN - 1 - s);
    v8f accA, accB;
    #pragma unroll
    for (int v = 0; v < 8; ++v) { accA[v] = bsum; accB[v] = bsum; }

    const float* hsrc = hbuf + (size_t)((cur * 2 + d) * BATCH) * HID;

    for (int kc = 0; kc < 16; ++kc) {        // 16 chunks x 64 K, double-buffered
      wait_async();                          // my slice of chunk kc is in LDS
      __syncthreads();                       // everyone's slice is in LDS
      if (kc < 15) {                         // overlap DMA of next chunk w/ WMMA
        const int kb1 = (kc + 1) * 64;
        const float* src1 = (kb1 < INP)
            ? (X + (size_t)t * BATCH * INP + kb1)
            : (hsrc + (kb1 - INP));
        issue_chunk_async(src1, ((kc + 1) & 1) ? a1off : a0off, wv, lane);
      }
      if (kc == 8 && s + 1 < S_LEN) {        // prefetch next timestep's X slab
        int tn = (d == 0) ? (s + 1) : (S_LEN - 2 - s);
        const char* pf = (const char*)(X + (size_t)tn * BATCH * INP);
        __builtin_prefetch(pf + tid * 512, 0, 1);
        __builtin_prefetch(pf + tid * 512 + 256, 0, 1);
      }
      compute_chunk(Wl, (kc & 1) ? act1 : act0, kc * 64, N0, M0a, M0b, l, hi,
                    lane, accA, accB);
    }

    // dump gates to LDS: C tile layout -> gld[m][q]  (gld aliases act0; act0's
    // last use was compute of chunk 14, fenced by the kc=15 __syncthreads)
    #pragma unroll
    for (int v = 0; v < 8; ++v) {
      int mr = v + (hi ? 8 : 0);
      gld[(M0a + mr) * GLD_STRIDE + N0 + l] = accA[v];
      gld[(M0b + mr) * GLD_STRIDE + N0 + l] = accB[v];
    }
    __syncthreads();

    // elementwise LSTM cell for this block's 16 columns x 64 batch
    {
      const int b = tid & 63;
      const int jb = (tid >> 6) * 4;
      const int sl = sls[b];
      float* Yrow = Y + (size_t)((t * 2 + d) * BATCH + b) * HID + j0;
      float* hdst = hbuf + (size_t)(((cur ^ 1) * 2 + d) * BATCH + b) * HID + j0;
      #pragma unroll
      for (int u = 0; u < 4; ++u) {
        int jj = jb + u;
        float gi = gld[b * GLD_STRIDE + jj];
        float go = gld[b * GLD_STRIDE + 16 + jj];
        float gf = gld[b * GLD_STRIDE + 32 + jj];
        float gc = gld[b * GLD_STRIDE + 48 + jj];
        float cold = cst[jj * 64 + b];
        float it = sigm(gi + Ps[jj] * cold);         // P_i
        float ft = sigm(gf + Ps[16 + jj] * cold);    // P_f
        float ctl = tanhf(gc);
        float cn = ft * cold + it * ctl;
        float ot = sigm(go + Ps[32 + jj] * cn);      // P_o
        float hn = ot * tanhf(cn);
        if (t >= sl) { hn = h0s[jj * 64 + b]; cn = c0s[jj * 64 + b]; }
        cst[jj * 64 + b] = cn;
        __builtin_nontemporal_store(hn, &Yrow[jj]);
        hdst[jj] = hn;
      }
    }
    cur ^= 1;

    __threadfence();
    __syncthreads();          // all gld reads done -> act0 reusable

    // hide next step's X DMA under the device-wide barrier: chunk 0 is pure X,
    // independent of other blocks' h
    if (s + 1 < S_LEN) {
      const int tn = (d == 0) ? (s + 1) : (S_LEN - 2 - s);
      issue_chunk_async(X + (size_t)tn * BATCH * INP, a0off, wv, lane);
    }

    // device-scope barrier across this direction's 32 blocks
    if (tid == 0) {
      unsigned target = (unsigned)(NBLK_DIR) * (unsigned)(s + 1);
      __hip_atomic_fetch_add(&cnt[d], 1u, __ATOMIC_RELEASE,
                             __HIP_MEMORY_SCOPE_AGENT);
      while (__hip_atomic_load(&cnt[d], __ATOMIC_ACQUIRE,
                               __HIP_MEMORY_SCOPE_AGENT) < target) {
        __builtin_amdgcn_s_sleep(1);
      }
    }
    __syncthreads();
  }

  // finals: Y_h from hbuf[cur], Y_c from LDS state
  {
    wait_async();   // no-op: the last step issues no async chunk
    const int b = tid & 63;
    const int jb = (tid >> 6) * 4;
    const float* hf = hbuf + (size_t)((cur * 2 + d) * BATCH + b) * HID + j0;
    float* Yh = Y + (size_t)S_LEN * 2 * BATCH * HID + (size_t)(d * BATCH + b) * HID + j0;
    float* Yc = Yh + (size_t)2 * BATCH * HID;
    #pragma unroll
    for (int u = 0; u < 4; ++u) {
      Yh[jb + u] = hf[jb + u];
      Yc[jb + u] = cst[(jb + u) * 64 + b];
    }
  }
}

extern "C" void kernel_launch(void* const* d_in, const int* in_sizes, int n_in,
                              void* d_out, int out_size, void* d_ws,
                              size_t ws_size, hipStream_t stream) {
  (void)in_sizes; (void)n_in; (void)out_size; (void)ws_size;
  const float* X  = (const float*)d_in[0];
  const float* W  = (const float*)d_in[1];
  const float* R  = (const float*)d_in[2];
  const float* Bg = (const float*)d_in[3];
  const int*   sq = (const int*)d_in[4];
  const float* h0 = (const float*)d_in[5];
  const float* c0 = (const float*)d_in[6];
  const float* P  = (const float*)d_in[7];
  float* Y = (float*)d_out;

  unsigned* cnt = (unsigned*)d_ws;                       // 2 counters
  float* hbuf = (float*)((char*)d_ws + 256);             // hbuf[2][D][B][H]

  (void)hipFuncSetAttribute((const void*)lstm_persist,
                            hipFuncAttributeMaxDynamicSharedMemorySize,
                            (int)LDS_BYTES);

  lstm_init<<<(2 * BATCH * HID) / THREADS, THREADS, 0, stream>>>(h0, hbuf, cnt);
  lstm_persist<<<2 * NBLK_DIR, THREADS, LDS_BYTES, stream>>>(
      X, W, R, Bg, sq, h0, c0, P, Y, hbuf, cnt);
}